// points2mult_image_55482387529810
// MI455X (gfx1250) — compile-verified
//
#include <hip/hip_runtime.h>
#include <stdint.h>

// Problem constants (from reference)
#define BOX   256
#define NC    10
#define BATCH 16
#define NPTS  262144
#define SS    (BOX * BOX)

// Tiling: 256 threads (8 wave32), 8 chunks of 256 points, double-buffered LDS.
#define THREADS 256
#define CHUNK   256
#define NITER   8
#define PPB     (CHUNK * NITER)   // 2048 points per block

#if defined(__has_builtin)
#  if __has_builtin(__builtin_amdgcn_tensor_load_to_lds)
#    define HAVE_TDM 1
#  else
#    define HAVE_TDM 0
#  endif
#else
#  define HAVE_TDM 0
#endif

typedef unsigned int u32x4 __attribute__((ext_vector_type(4)));
typedef int          i32x4 __attribute__((ext_vector_type(4)));
typedef int          i32x8 __attribute__((ext_vector_type(8)));

__device__ __forceinline__ uint32_t lds_lo32(const void* p) {
    // Low 32 bits of a generic LDS-aperture pointer == workgroup-relative LDS
    // byte offset (ISA aperture mapping) — what async/TDM LDS operands expect.
    return (uint32_t)(uintptr_t)p;
}

// CDNA5 async global->LDS (ASYNCcnt). GV mode: vdst=LDS byte addr, vaddr=64b.
__device__ __forceinline__ void async_load_f32(const float* g, uint32_t lds_byte) {
    asm volatile("global_load_async_to_lds_b32 %0, %1, off"
                 :: "v"(lds_byte), "v"(g) : "memory");
}

// Fire-and-forget f32 atomic add (relaxed, device scope -> global_atomic_add_f32,
// no return => STOREcnt, no dependency back into the wave).
__device__ __forceinline__ void fadd_nr(float* p, float v) {
    (void)__hip_atomic_fetch_add(p, v, __ATOMIC_RELAXED, __HIP_MEMORY_SCOPE_AGENT);
}

#if HAVE_TDM
// One TDM 2D tile load: tile_d1 rows x tile_d0 f32 elements, row stride
// stride0 (elements), from tile start `gaddr` into LDS at lds_byte.
// D# layout per CDNA5 ISA 8.3/8.4 (group0 128b, group1 256b, groups 2/3 zero).
__device__ __forceinline__ void tdm_load_tile_f32(
    uint32_t lds_byte, const float* gaddr,
    uint32_t tensor_d0, uint32_t tensor_d1,
    uint32_t tile_d0, uint32_t tile_d1, uint64_t stride0)
{
    const uint64_t ga = (uint64_t)(uintptr_t)gaddr;
    u32x4 g0;
    g0.x = 1u;                                            // count=1, user, no gather
    g0.y = lds_byte;                                      // lds_addr
    g0.z = (uint32_t)ga;                                  // global_addr[31:0]
    g0.w = (uint32_t)((ga >> 32) & 0x01FFFFFFu)           // global_addr[56:32]
         | (2u << 30);                                    // type = 2 ("image")
    i32x8 g1;
    g1[0] = (int)(2u << 16);                              // data_size=2 (4B); mask/flags=0
    g1[1] = (int)((tensor_d0 & 0xFFFFu) << 16);           // dim0[15:0]   (bits 63:48)
    g1[2] = (int)(((tensor_d0 >> 16) & 0xFFFFu)           // dim0[31:16]  (bits 79:64)
                | ((tensor_d1 & 0xFFFFu) << 16));         // dim1[15:0]   (bits 95:80)
    g1[3] = (int)(((tensor_d1 >> 16) & 0xFFFFu)           // dim1[31:16]  (bits 111:96)
                | ((tile_d0 & 0xFFFFu) << 16));           // tile_dim0    (bits 127:112)
    g1[4] = (int)(tile_d1 & 0xFFFFu);                     // tile_dim1; tile_dim2=0
    g1[5] = (int)(uint32_t)(stride0 & 0xFFFFFFFFu);       // stride0[31:0]
    g1[6] = (int)(uint32_t)((stride0 >> 32) & 0xFFFFu);   // stride0[47:32]; stride1=0
    g1[7] = 0;
    i32x4 gz = {0, 0, 0, 0};                              // groups 2/3: dims/tiles unused
#if __has_include(<hip/amd_detail/amd_gfx1250_TDM.h>)
    i32x8 gz8 = {0, 0, 0, 0, 0, 0, 0, 0};                 // clang-23 6-arg form
    __builtin_amdgcn_tensor_load_to_lds(g0, g1, gz, gz, gz8, 0);
#else
    __builtin_amdgcn_tensor_load_to_lds(g0, g1, gz, gz, 0); // ROCm 7.2 5-arg form
#endif
}
#endif // HAVE_TDM

__global__ __launch_bounds__(THREADS) void zero_out_kernel(float4* __restrict__ out, int n4) {
    int i = blockIdx.x * blockDim.x + threadIdx.x;
    if (i < n4) out[i] = make_float4(0.f, 0.f, 0.f, 0.f);
}

__global__ __launch_bounds__(THREADS) void cic_scatter_kernel(
    const float* __restrict__ points,   // [B, 2, N]
    const float* __restrict__ values,   // [B, NC, N]
    float* __restrict__ out)            // [B, NC, 256, 256]
{
    __shared__ float vbuf[2][NC * CHUNK];   // 20 KB: values tiles (TDM dest)
    __shared__ float pbuf[2][2 * CHUNK];    //  4 KB: x/y coords (async dest)

    const int tid = threadIdx.x;
    const int blocksPerBatch = NPTS / PPB;                 // 128
    const int b    = blockIdx.x / blocksPerBatch;
    const int base = (blockIdx.x % blocksPerBatch) * PPB;

    const float* px = points + (size_t)b * 2 * NPTS;
    const float* py = px + NPTS;
    const float* vb = values + (size_t)b * NC * NPTS;
    float*       ob = out    + (size_t)b * NC * SS;

    const uint32_t vlds[2] = { lds_lo32(&vbuf[0][0]), lds_lo32(&vbuf[1][0]) };
    const uint32_t plds[2] = { lds_lo32(&pbuf[0][0]), lds_lo32(&pbuf[1][0]) };

    // ---- prologue: stage chunk 0 -----------------------------------------
    async_load_f32(px + base + tid, plds[0] + (uint32_t)tid * 4u);
    async_load_f32(py + base + tid, plds[0] + (uint32_t)(CHUNK + tid) * 4u);
#if HAVE_TDM
    if (tid < 32)   // one wave issues the DMA; TENSORcnt is per-wave
        tdm_load_tile_f32(vlds[0], vb + base, NPTS, NC, CHUNK, NC, NPTS);
#else
    {
        const float* g = vb + base + tid;
#pragma unroll
        for (int c = 0; c < NC; ++c)
            async_load_f32(g + (size_t)c * NPTS, vlds[0] + (uint32_t)(c * CHUNK + tid) * 4u);
    }
#endif

    for (int it = 0; it < NITER; ++it) {
        const int buf = it & 1;

        // ---- issue chunk it+1, then wait for chunk it (in-order done) ----
        if (it + 1 < NITER) {
            const int nb = buf ^ 1;
            const int p1 = base + (it + 1) * CHUNK;
            async_load_f32(px + p1 + tid, plds[nb] + (uint32_t)tid * 4u);
            async_load_f32(py + p1 + tid, plds[nb] + (uint32_t)(CHUNK + tid) * 4u);
#if HAVE_TDM
            if (tid < 32)
                tdm_load_tile_f32(vlds[nb], vb + p1, NPTS, NC, CHUNK, NC, NPTS);
            asm volatile("s_wait_asynccnt %0" :: "n"(2) : "memory");
            if (tid < 32) __builtin_amdgcn_s_wait_tensorcnt(1);
#else
            {
                const float* g = vb + p1 + tid;
#pragma unroll
                for (int c = 0; c < NC; ++c)
                    async_load_f32(g + (size_t)c * NPTS,
                                   vlds[nb] + (uint32_t)(c * CHUNK + tid) * 4u);
            }
            asm volatile("s_wait_asynccnt %0" :: "n"(12) : "memory");
#endif
        } else {
            asm volatile("s_wait_asynccnt %0" :: "n"(0) : "memory");
#if HAVE_TDM
            if (tid < 32) __builtin_amdgcn_s_wait_tensorcnt(0);
#endif
        }
        __syncthreads();   // TDM wrote LDS via wave 0: make visible to all waves

        // ---- bilinear weights for this thread's point --------------------
        const float pxv = (pbuf[buf][tid] + 0.5f) * (float)BOX;
        const float pyv = (pbuf[buf][CHUNK + tid] + 0.5f) * (float)BOX;
        const float xf = floorf(pxv);
        const float yf = floorf(pyv);
        const float rx = pxv - xf;
        const float ry = pyv - yf;
        const int ix = (int)xf;
        const int iy = (int)yf;

        const bool okx0 = (ix >= 0) && (ix < BOX);
        const bool oky0 = (iy >= 0) && (iy < BOX);
        const bool okx1 = (ix + 1 >= 0) && (ix + 1 < BOX);
        const bool oky1 = (iy + 1 >= 0) && (iy + 1 < BOX);

        const float wx0 = 1.0f - rx, wx1 = rx;
        const float wy0 = 1.0f - ry, wy1 = ry;
        const float w00 = wx0 * wy0, w10 = wx1 * wy0;
        const float w01 = wx0 * wy1, w11 = wx1 * wy1;

        const int i00 = iy * BOX + ix;
        const float* lv = &vbuf[buf][tid];

#pragma unroll
        for (int c = 0; c < NC; ++c) {
            const float v = lv[c * CHUNK];          // ds_load_b32 from staged tile
            float* im = ob + (size_t)c * SS;
            if (okx0 && oky0) fadd_nr(im + i00,           w00 * v);
            if (okx1 && oky0) fadd_nr(im + i00 + 1,       w10 * v);
            if (okx0 && oky1) fadd_nr(im + i00 + BOX,     w01 * v);
            if (okx1 && oky1) fadd_nr(im + i00 + BOX + 1, w11 * v);
        }
        __syncthreads();   // all waves done reading buf before it is refilled
    }
}

extern "C" void kernel_launch(void* const* d_in, const int* in_sizes, int n_in,
                              void* d_out, int out_size, void* d_ws, size_t ws_size,
                              hipStream_t stream) {
    const float* points = (const float*)d_in[0];   // [16, 2, 262144] f32
    const float* values = (const float*)d_in[1];   // [16, 10, 262144] f32
    float* out = (float*)d_out;                    // [16, 10, 256, 256] f32

    const int n4 = out_size / 4;
    zero_out_kernel<<<(n4 + THREADS - 1) / THREADS, THREADS, 0, stream>>>(
        (float4*)out, n4);

    const int blocks = (BATCH * NPTS) / PPB;       // 2048 blocks, 8 wave32 each
    cic_scatter_kernel<<<blocks, THREADS, 0, stream>>>(points, values, out);
}